// EGATLayer_70153995813493
// MI455X (gfx1250) — compile-verified
//
#include <hip/hip_runtime.h>
#include <hip/hip_bf16.h>

#define NEG_SLOPE 0.2f

typedef __attribute__((ext_vector_type(2))) float v2f;
typedef __attribute__((ext_vector_type(4))) float v4f;
typedef __attribute__((ext_vector_type(8))) float v8f;

// Monotonic float<->uint key so segment-max becomes a single u32 atomic max.
__device__ __forceinline__ unsigned fkey(float x) {
    unsigned b = __float_as_uint(x);
    return (b & 0x80000000u) ? ~b : (b | 0x80000000u);
}
__device__ __forceinline__ float funkey(unsigned k) {
    unsigned b = (k & 0x80000000u) ? (k ^ 0x80000000u) : ~k;
    return __uint_as_float(b);
}

// ---------------- init: emax keys / denom ----------------
__global__ void init_kernel(unsigned* __restrict__ emax_keys,
                            float* __restrict__ denom, int N) {
    int i = blockIdx.x * blockDim.x + threadIdx.x;
    if (i < N) { emax_keys[i] = 0u; denom[i] = 0.0f; }
}

// ---------------- b_e = W_edge @ a_e (64x64 @ 64) ----------------
__global__ void be_kernel(const float* __restrict__ W_edge,
                          const float* __restrict__ a_vec,
                          float* __restrict__ b_e) {
    int i = threadIdx.x; // 64 threads
    float s = 0.f;
#pragma unroll
    for (int j = 0; j < 64; ++j) s = fmaf(W_edge[i * 64 + j], a_vec[128 + j], s);
    b_e[i] = s;
}

// ---------------- node GEMMs via V_WMMA_F32_16X16X4_F32 ----------------
// Each wave owns a 16-row x 64-col output tile of BOTH Wh and Hres
// (the A fragment is shared between the two GEMMs). Block = 8 waves = 128 rows.
__global__ __launch_bounds__(256) void node_gemm_kernel(
    const float* __restrict__ X,   // N x 128
    const float* __restrict__ Wn,  // 128 x 64
    const float* __restrict__ Wr,  // 128 x 64
    float* __restrict__ Wh,        // N x 64
    float* __restrict__ Hres,      // N x 64 (d_out)
    int N) {
    const int lane   = threadIdx.x & 31;
    const int wave   = threadIdx.x >> 5;
    const int hi     = lane >> 4;      // 0: lanes 0-15, 1: lanes 16-31
    const int lanelo = lane & 15;
    const int row0   = blockIdx.x * 128 + wave * 16;

    int rA = row0 + lanelo;
    if (rA >= N) rA = N - 1;           // clamp loads; stores are guarded

    v8f accn[4], accr[4];
#pragma unroll
    for (int j = 0; j < 4; ++j) { accn[j] = (v8f){}; accr[j] = (v8f){}; }

    const float* arow = X + (long)rA * 128 + 2 * hi;

#pragma unroll 4
    for (int k = 0; k < 128; k += 4) {
        // A 16x4 fragment: lane half selects K pair, 2 VGPRs hold the pair
        v2f a = *(const v2f*)(arow + k);
#pragma unroll
        for (int j = 0; j < 4; ++j) {
            const int col = 16 * j + lanelo;
            v2f bn, br;
            bn.x = Wn[(k + 2 * hi) * 64 + col];
            bn.y = Wn[(k + 2 * hi + 1) * 64 + col];
            br.x = Wr[(k + 2 * hi) * 64 + col];
            br.y = Wr[(k + 2 * hi + 1) * 64 + col];
            accn[j] = __builtin_amdgcn_wmma_f32_16x16x4_f32(
                false, a, false, bn, (short)0, accn[j], false, false);
            accr[j] = __builtin_amdgcn_wmma_f32_16x16x4_f32(
                false, a, false, br, (short)0, accr[j], false, false);
        }
    }

    // C/D layout: VGPR v, lane L -> row = v + 8*(L>=16), col = (L&15) + 16j
#pragma unroll
    for (int j = 0; j < 4; ++j) {
#pragma unroll
        for (int v = 0; v < 8; ++v) {
            const int row = row0 + v + 8 * hi;
            const int col = 16 * j + lanelo;
            if (row < N) {
                Wh[(long)row * 64 + col]   = accn[j][v];
                Hres[(long)row * 64 + col] = accr[j][v];
            }
        }
    }
}

// ---------------- per-node attention scalars ----------------
__global__ void svec_kernel(const float* __restrict__ Wh,
                            const float* __restrict__ a_vec,
                            float* __restrict__ s_src,
                            float* __restrict__ s_dst, int N) {
    int n = blockIdx.x * blockDim.x + threadIdx.x;
    if (n >= N) return;
    const v4f* row = (const v4f*)(Wh + (long)n * 64);
    const v4f* as  = (const v4f*)(a_vec);
    const v4f* ad  = (const v4f*)(a_vec + 64);
    float ss = 0.f, sd = 0.f;
#pragma unroll
    for (int i = 0; i < 16; ++i) {
        v4f r = row[i], x = as[i], y = ad[i];
        ss += r.x * x.x + r.y * x.y + r.z * x.z + r.w * x.w;
        sd += r.x * y.x + r.y * y.y + r.z * y.z + r.w * y.w;
    }
    s_src[n] = ss;
    s_dst[n] = sd;
}

// ---------------- per-edge score + LeakyReLU + segment max ----------------
__global__ void edge_score_kernel(const float* __restrict__ EF,
                                  const float* __restrict__ b_e,
                                  const float* __restrict__ s_src,
                                  const float* __restrict__ s_dst,
                                  const int* __restrict__ src,
                                  const int* __restrict__ dst,
                                  float* __restrict__ e_buf,
                                  unsigned* __restrict__ emax_keys, int E) {
    int e = blockIdx.x * blockDim.x + threadIdx.x;
    if (e >= E) return;
    const v4f* ef = (const v4f*)(EF + (long)e * 64);
    const v4f* be = (const v4f*)b_e;
    float acc = 0.f;
#pragma unroll
    for (int i = 0; i < 16; ++i) {
        v4f r = __builtin_nontemporal_load(ef + i);  // 410 MB stream: TH=NT
        v4f b = be[i];
        acc += r.x * b.x + r.y * b.y + r.z * b.z + r.w * b.w;
    }
    int s = src[e], d = dst[e];
    float val = acc + s_src[s] + s_dst[d];
    val = (val >= 0.f) ? val : NEG_SLOPE * val;
    e_buf[e] = val;
    atomicMax(&emax_keys[d], fkey(val));
}

// ---------------- exp + segment sum ----------------
__global__ void edge_exp_kernel(const float* __restrict__ e_buf,
                                const unsigned* __restrict__ emax_keys,
                                const int* __restrict__ dst,
                                float* __restrict__ ex_buf,
                                float* __restrict__ denom, int E) {
    int e = blockIdx.x * blockDim.x + threadIdx.x;
    if (e >= E) return;
    int d = dst[e];
    float ex = __expf(e_buf[e] - funkey(emax_keys[d]));
    ex_buf[e] = ex;
    unsafeAtomicAdd(&denom[d], ex);   // native global_atomic_add_f32
}

// ---------------- alpha = ex / denom[dst] (in place) ----------------
__global__ void alpha_kernel(float* __restrict__ ex_buf,
                             const float* __restrict__ denom,
                             const int* __restrict__ dst, int E) {
    int e = blockIdx.x * blockDim.x + threadIdx.x;
    if (e >= E) return;
    ex_buf[e] = ex_buf[e] / denom[dst[e]];
}

// ---------------- message scatter: out[dst] += Wh[src] * alpha ----------------
// 64 threads per edge (one per output channel); edge is wave-uniform.
__global__ __launch_bounds__(256) void scatter_kernel(
    const float* __restrict__ Wh, const float* __restrict__ alpha,
    const int* __restrict__ src, const int* __restrict__ dst,
    float* __restrict__ out, int E) {
    unsigned gid = blockIdx.x * 256u + threadIdx.x;
    unsigned e = gid >> 6;
    unsigned c = gid & 63u;
    if (e >= (unsigned)E) return;
    int s = src[e], d = dst[e];
    float a = alpha[e];
    unsafeAtomicAdd(&out[(long)d * 64 + c], Wh[(long)s * 64 + c] * a);
}

// ---------------- final ReLU ----------------
__global__ void relu_kernel(float* __restrict__ out, int n) {
    int i = blockIdx.x * blockDim.x + threadIdx.x;
    if (i < n) {
        float x = out[i];
        out[i] = x > 0.f ? x : 0.f;
    }
}

extern "C" void kernel_launch(void* const* d_in, const int* in_sizes, int n_in,
                              void* d_out, int out_size, void* d_ws, size_t ws_size,
                              hipStream_t stream) {
    const float* node_feats = (const float*)d_in[0];
    const float* edge_feats = (const float*)d_in[1];
    const float* W_node     = (const float*)d_in[2];
    const float* W_edge     = (const float*)d_in[3];
    const float* a_vec      = (const float*)d_in[4];
    const float* W_res      = (const float*)d_in[5];
    const int*   src        = (const int*)d_in[6];
    const int*   dst        = (const int*)d_in[7];
    float*       out        = (float*)d_out;

    const int N = in_sizes[0] / 128;
    const int E = in_sizes[6];

    // Workspace layout (256B aligned regions)
    char* ws = (char*)d_ws;
    auto align256 = [](size_t x) { return (x + 255) & ~(size_t)255; };
    size_t off = 0;
    float*    Wh        = (float*)(ws + off); off = align256(off + (size_t)N * 64 * 4);
    float*    s_src     = (float*)(ws + off); off = align256(off + (size_t)N * 4);
    float*    s_dst     = (float*)(ws + off); off = align256(off + (size_t)N * 4);
    unsigned* emax_keys = (unsigned*)(ws + off); off = align256(off + (size_t)N * 4);
    float*    denom     = (float*)(ws + off); off = align256(off + (size_t)N * 4);
    float*    e_buf     = (float*)(ws + off); off = align256(off + (size_t)E * 4);
    float*    ex_buf    = (float*)(ws + off); off = align256(off + (size_t)E * 4);
    float*    b_e       = (float*)(ws + off); off = align256(off + 64 * 4);

    init_kernel<<<(N + 255) / 256, 256, 0, stream>>>(emax_keys, denom, N);
    be_kernel<<<1, 64, 0, stream>>>(W_edge, a_vec, b_e);
    node_gemm_kernel<<<(N + 127) / 128, 256, 0, stream>>>(node_feats, W_node, W_res,
                                                          Wh, out, N);
    svec_kernel<<<(N + 255) / 256, 256, 0, stream>>>(Wh, a_vec, s_src, s_dst, N);
    edge_score_kernel<<<(E + 255) / 256, 256, 0, stream>>>(edge_feats, b_e, s_src, s_dst,
                                                           src, dst, e_buf, emax_keys, E);
    edge_exp_kernel<<<(E + 255) / 256, 256, 0, stream>>>(e_buf, emax_keys, dst,
                                                         ex_buf, denom, E);
    alpha_kernel<<<(E + 255) / 256, 256, 0, stream>>>(ex_buf, denom, dst, E);
    scatter_kernel<<<(int)(((size_t)E * 64 + 255) / 256), 256, 0, stream>>>(
        Wh, ex_buf, src, dst, out, E);
    relu_kernel<<<(N * 64 + 255) / 256, 256, 0, stream>>>(out, N * 64);
}